// GCN_NODE_WEIGHT_14104672600539
// MI455X (gfx1250) — compile-verified
//
#include <hip/hip_runtime.h>

typedef float v2f __attribute__((ext_vector_type(2)));
typedef float v4f __attribute__((ext_vector_type(4)));
typedef float v8f __attribute__((ext_vector_type(8)));

#define IN_D   256
#define OUT_D  256
#define KNB    32
#define M_BLK  64
#define LDS_PITCH (IN_D + 4)   // 260 floats: row stride ≡ 4 mod 64 banks -> conflict-free

// ---------------------------------------------------------------------------
// Pack Wc|Wn into k-pair-interleaved float2: Wp[mat][k/2][n] = {W[k][n], W[k+1][n]}
// Makes every WMMA B-fragment a single b64 load. 65536 elements total.
// ---------------------------------------------------------------------------
__global__ __launch_bounds__(256)
void pack_w(const float* __restrict__ Wc, const float* __restrict__ Wn,
            v2f* __restrict__ Wp)
{
    int i   = blockIdx.x * 256 + threadIdx.x;   // 0 .. 2*128*256-1
    int mat = i >> 15;
    int r   = i & 32767;
    int k2  = r >> 8;
    int n   = r & 255;
    const float* __restrict__ W = mat ? Wn : Wc;
    v2f v;
    v.x = W[(size_t)(2 * k2) * OUT_D + n];
    v.y = W[(size_t)(2 * k2 + 1) * OUT_D + n];
    Wp[i] = v;
}

// ---------------------------------------------------------------------------
// GEMM: Y[m0:m0+64, :] = X[m0:m0+64, :] @ W   via V_WMMA_F32_16X16X4_F32
// 256 threads = 8 waves. Wave w owns N-tiles {2w, 2w+1} and ALL 4 M-tiles,
// so each W element is loaded exactly once per block (min L2 traffic).
// gridDim.y: 0 -> Wc, Y=Zc(d_out); 1 -> Wn, Y=vWn(ws).
// ---------------------------------------------------------------------------
__global__ __launch_bounds__(256)
void gemm_f32_wmma(const float* __restrict__ X, const v2f* __restrict__ Wp,
                   float* __restrict__ Yc, float* __restrict__ Yn, int nrows)
{
    __shared__ float sx[M_BLK * LDS_PITCH];     // 64*260*4 = 66,560 B

    const int tid = threadIdx.x;
    const int m0  = blockIdx.x * M_BLK;
    const v2f* __restrict__ Wm = Wp + (size_t)blockIdx.y * (IN_D / 2) * OUT_D;
    float* __restrict__ Y = blockIdx.y ? Yn : Yc;

    // Stage the 64x256 X tile (float4, padded pitch). 4096 float4 / 256 thr.
    {
        const v4f* __restrict__ X4 = reinterpret_cast<const v4f*>(X);
        v4f* s4 = reinterpret_cast<v4f*>(sx);
        const v4f zero = {0.f, 0.f, 0.f, 0.f};
        #pragma unroll
        for (int t = 0; t < (M_BLK * IN_D / 4) / 256; ++t) {
            int li   = tid + t * 256;
            int row  = li >> 6;
            int c4   = li & 63;
            int grow = m0 + row;
            v4f v = (grow < nrows) ? X4[(size_t)grow * (IN_D / 4) + c4] : zero;
            s4[row * (LDS_PITCH / 4) + c4] = v;
        }
    }
    __syncthreads();

    const int w    = tid >> 5;
    const int lane = tid & 31;
    const int mrow = lane & 15;            // A: M index of this lane
    const int kk   = (lane >> 4) << 1;     // A/B: K sub-offset (0 or 2)
    const int ncol = lane & 15;            // B/C/D: N index of this lane
    const int c0   = (w * 2 + 0) * 16 + ncol;
    const int c1   = (w * 2 + 1) * 16 + ncol;

    v8f acc[4][2] = {};

    #pragma unroll 2
    for (int k = 0; k < IN_D; k += 4) {
        const int krow = (k + kk) >> 1;                    // packed-pair row
        v2f b0 = Wm[(size_t)krow * OUT_D + c0];            // single b64 load
        v2f b1 = Wm[(size_t)krow * OUT_D + c1];
        #pragma unroll
        for (int m = 0; m < 4; ++m) {
            const float* __restrict__ sA = &sx[(m * 16 + mrow) * LDS_PITCH + k + kk];
            v2f a;
            a.x = sA[0];
            a.y = sA[1];
            acc[m][0] = __builtin_amdgcn_wmma_f32_16x16x4_f32(false, a, false, b0, (short)0, acc[m][0], false, false);
            acc[m][1] = __builtin_amdgcn_wmma_f32_16x16x4_f32(false, a, false, b1, (short)0, acc[m][1], false, false);
        }
    }

    // C/D layout: VGPR r -> M = r (lanes 0-15) / r+8 (lanes 16-31), N = lane&15.
    const int rsub = (lane >> 4) << 3;
    if (m0 + M_BLK <= nrows) {
        // Fast path (all blocks but the last): one base address, all stores
        // use immediate offsets (max 3*16KB + 7*1KB + 64B < 24-bit IOFFSET).
        float* __restrict__ base = Y + (size_t)(m0 + rsub) * OUT_D + c0;
        #pragma unroll
        for (int m = 0; m < 4; ++m) {
            #pragma unroll
            for (int r = 0; r < 8; ++r) {
                base[(m * 16 + r) * OUT_D]      = acc[m][0][r];
                base[(m * 16 + r) * OUT_D + 16] = acc[m][1][r];
            }
        }
    } else {
        #pragma unroll
        for (int m = 0; m < 4; ++m) {
            #pragma unroll
            for (int r = 0; r < 8; ++r) {
                int row = m0 + m * 16 + rsub + r;
                if (row < nrows) {
                    float* __restrict__ yr = Y + (size_t)row * OUT_D;
                    yr[c0] = acc[m][0][r];
                    yr[c1] = acc[m][1][r];
                }
            }
        }
    }
}

// ---------------------------------------------------------------------------
// Pass 2: one wave per node.
//   Zn = (Σ_k vWn[adj[n,k]] + es0*We[0] + es1*We[1]) / max(nh,1)
//   out = relu(Zn + Zc + b)        (Zc already resident in d_out)
// Softmax over a size-1 axis == 1, so q is dead and weights are uniform.
// ---------------------------------------------------------------------------
__global__ __launch_bounds__(256)
void gather_finalize(const float* __restrict__ vWn,
                     const int*   __restrict__ adj,
                     const float* __restrict__ edge,
                     const float* __restrict__ We,
                     const float* __restrict__ bias,
                     float* out, int nrows)
{
    const int lane = threadIdx.x & 31;
    const int w    = threadIdx.x >> 5;
    const int n    = blockIdx.x * 8 + w;
    if (n >= nrows) return;

    // Lane k owns neighbor k.
    int idx = adj[(size_t)n * KNB + lane];
    float nv = (idx != -1) ? 1.0f : 0.0f;      // count_nonzero(adj+1)
    if (idx < 0) idx += nrows;                 // JAX negative-index wrap

    v2f e = *reinterpret_cast<const v2f*>(edge + ((size_t)n * KNB + lane) * 2);
    float es0 = e.x, es1 = e.y;

    #pragma unroll
    for (int off = 16; off; off >>= 1) {
        es0 += __shfl_xor(es0, off, 32);
        es1 += __shfl_xor(es1, off, 32);
        nv  += __shfl_xor(nv,  off, 32);
    }
    const float inv = 1.0f / fmaxf(nv, 1.0f);

    // Lane covers channels [lane*8, lane*8+8): two float4's (1 KB/row, coalesced).
    const int d4 = lane * 2;
    v4f acc0 = {}, acc1 = {};
    #pragma unroll 4
    for (int k = 0; k < KNB; ++k) {
        int g = __shfl(idx, k, 32);
        const v4f* __restrict__ r = reinterpret_cast<const v4f*>(vWn + (size_t)g * OUT_D);
        acc0 += r[d4];
        acc1 += r[d4 + 1];
    }

    const v4f* __restrict__ We0 = reinterpret_cast<const v4f*>(We);
    const v4f* __restrict__ We1 = reinterpret_cast<const v4f*>(We + OUT_D);
    const v4f* __restrict__ B4  = reinterpret_cast<const v4f*>(bias);
    v4f*                    O4  = reinterpret_cast<v4f*>(out);

    v4f zc0 = O4[(size_t)n * (OUT_D / 4) + d4];
    v4f zc1 = O4[(size_t)n * (OUT_D / 4) + d4 + 1];

    v4f z0 = (acc0 + es0 * We0[d4]     + es1 * We1[d4])     * inv + zc0 + B4[d4];
    v4f z1 = (acc1 + es0 * We0[d4 + 1] + es1 * We1[d4 + 1]) * inv + zc1 + B4[d4 + 1];

    v4f r0, r1;
    #pragma unroll
    for (int j = 0; j < 4; ++j) {
        r0[j] = fmaxf(z0[j], 0.0f);
        r1[j] = fmaxf(z1[j], 0.0f);
    }
    O4[(size_t)n * (OUT_D / 4) + d4]     = r0;
    O4[(size_t)n * (OUT_D / 4) + d4 + 1] = r1;
}

// ---------------------------------------------------------------------------
extern "C" void kernel_launch(void* const* d_in, const int* in_sizes, int n_in,
                              void* d_out, int out_size, void* d_ws, size_t ws_size,
                              hipStream_t stream)
{
    const float* x    = (const float*)d_in[0];
    const int*   adj  = (const int*)  d_in[1];
    const float* edge = (const float*)d_in[2];
    const float* Wc   = (const float*)d_in[3];
    const float* Wn   = (const float*)d_in[4];
    const float* We   = (const float*)d_in[5];
    // d_in[6] = q : unused (softmax over a size-1 axis is identically 1)
    const float* b    = (const float*)d_in[7];
    // d_in[8] = training : inference path, ignored

    const int nrows = in_sizes[0] / IN_D;        // N
    float* out = (float*)d_out;
    float* vWn = (float*)d_ws;                   // N*256 f32 scratch (51.2 MB)
    v2f*   Wp  = (v2f*)((char*)d_ws + (size_t)nrows * OUT_D * sizeof(float));

    pack_w<<<dim3(256), dim3(256), 0, stream>>>(Wc, Wn, Wp);

    dim3 g1((nrows + M_BLK - 1) / M_BLK, 2);
    gemm_f32_wmma<<<g1, dim3(256), 0, stream>>>(x, Wp, out, vWn, nrows);

    dim3 g2((nrows + 7) / 8);
    gather_finalize<<<g2, dim3(256), 0, stream>>>(vWn, adj, edge, We, b, out, nrows);
}